// Matcher_52862457479214
// MI455X (gfx1250) — compile-verified
//
#include <hip/hip_runtime.h>
#include <hip/hip_bf16.h>
#include <stdint.h>

#define N_BOX 4096
#define PI_F  3.141592653f
#define IOU_T 0.1f

typedef __attribute__((ext_vector_type(2))) float        v2f;
typedef __attribute__((ext_vector_type(8))) float        v8f;
typedef __attribute__((ext_vector_type(4))) unsigned int u32x4;
typedef __attribute__((ext_vector_type(4))) int          i32x4;
typedef __attribute__((ext_vector_type(8))) int          i32x8;

__device__ __forceinline__ float limit_period_f(float v) {
  return v - floorf(v / (2.f * PI_F) + 0.5f) * (2.f * PI_F);
}

// ---------------------------------------------------------------- prep0
__global__ void prep0_kernel(const float* __restrict__ boxes,
                             float* dirs, float* s_sum, int* cnt,
                             unsigned long long* refpack) {
  int i = blockIdx.x * blockDim.x + threadIdx.x;
  if (i >= N_BOX) return;
  dirs[i]    = limit_period_f(boxes[i * 7 + 6]);
  s_sum[i]   = 0.f;
  cnt[i]     = 0;
  refpack[i] = 0ull;
}

// ------------------------------------------------- pairwise IoU bitmask
// mask[row][word w] bit b  <=>  iou(row, w*32+b) > 0.1   (axis-aligned 3D)
__global__ void iou_mask_kernel(const float* __restrict__ boxes,
                                uint32_t* __restrict__ mask) {
  int row = blockIdx.x;
  int t   = threadIdx.x;  // 0..127, covers 32 columns each
  float rcx = boxes[row * 7 + 0], rcy = boxes[row * 7 + 1], rcz = boxes[row * 7 + 2];
  float rh  = boxes[row * 7 + 3], rw  = boxes[row * 7 + 4], rl  = boxes[row * 7 + 5];
  float rvol = rl * rw * rh;
  float rlox = rcx - 0.5f * rl, rhix = rcx + 0.5f * rl;
  float rloy = rcy - 0.5f * rw, rhiy = rcy + 0.5f * rw;
  float rloz = rcz - 0.5f * rh, rhiz = rcz + 0.5f * rh;
  uint32_t bits = 0;
  for (int b = 0; b < 32; ++b) {
    int j = t * 32 + b;
    float cx = boxes[j * 7 + 0], cy = boxes[j * 7 + 1], cz = boxes[j * 7 + 2];
    float h  = boxes[j * 7 + 3], w  = boxes[j * 7 + 4], l  = boxes[j * 7 + 5];
    float ix = fminf(rhix, cx + 0.5f * l) - fmaxf(rlox, cx - 0.5f * l);
    float iy = fminf(rhiy, cy + 0.5f * w) - fmaxf(rloy, cy - 0.5f * w);
    float iz = fminf(rhiz, cz + 0.5f * h) - fmaxf(rloz, cz - 0.5f * h);
    ix = fmaxf(ix, 0.f); iy = fmaxf(iy, 0.f); iz = fmaxf(iz, 0.f);
    float iv  = ix * iy * iz;
    float vol = l * w * h;
    float iou = iv / fmaxf(rvol + vol - iv, 1e-8f);
    if (iou > IOU_T) bits |= (1u << b);
  }
  mask[row * 128 + t] = bits;
}

// -------------------------------------- sequential greedy cluster assign
__global__ void assign_kernel(const uint32_t* __restrict__ mask,
                              int* __restrict__ indices) {
  __shared__ int sidx[N_BOX];
  __shared__ int ctrl[2];  // [0]=active for this i, [1]=next cluster id
  int t = threadIdx.x;     // 1024 threads, 4 columns each
  for (int j = t; j < N_BOX; j += 1024) sidx[j] = 0;
  if (t == 0) ctrl[1] = 1;
  __syncthreads();
  for (int i = 0; i < N_BOX; ++i) {
    if (t == 0) ctrl[0] = (sidx[i] == 0);
    __syncthreads();
    if (ctrl[0]) {                       // uniform branch across the block
      int cid = ctrl[1];
      uint32_t w   = mask[i * 128 + (t >> 3)];
      uint32_t nib = (w >> ((t & 7) * 4)) & 0xFu;
      int base = t * 4;
      if (nib & 1u) sidx[base + 0] = cid;
      if (nib & 2u) sidx[base + 1] = cid;
      if (nib & 4u) sidx[base + 2] = cid;
      if (nib & 8u) sidx[base + 3] = cid;
      __syncthreads();
      if (t == 0) ctrl[1] = cid + 1;
    }
    __syncthreads();
  }
  for (int j = t; j < N_BOX; j += 1024) indices[j] = sidx[j];
}

// ------------------------------------------ per-cluster stats (atomics)
__global__ void stats_kernel(const float* __restrict__ scores,
                             const int* __restrict__ indices,
                             float* s_sum, int* cnt, float* clscores,
                             unsigned long long* refpack) {
  int j = blockIdx.x * blockDim.x + threadIdx.x;
  if (j >= N_BOX) return;
  int idx = indices[j];
  if (idx <= 0) return;
  int r   = idx - 1;
  float s = scores[j];
  atomicAdd(&s_sum[r], s);
  int slot = atomicAdd(&cnt[r], 1);
  if (slot < 64) clscores[r * 64 + slot] = s;
  // argmax(score) with first-index tie-break: scores > 0 so float bits are monotone
  unsigned long long pack = (((unsigned long long)__float_as_uint(s)) << 32) |
                            (unsigned long long)(0xFFFFFFFFu - (unsigned)j);
  atomicMax(&refpack[r], pack);
}

// -------------------- per-box A-columns (score, signed score) + B panel
__global__ void prep2_kernel(const float* __restrict__ boxes,
                             const float* __restrict__ scores,
                             const float* __restrict__ dirs,
                             const int* __restrict__ indices,
                             const unsigned long long* __restrict__ refpack,
                             float2* __restrict__ colsA,
                             float* __restrict__ Bmat) {
  int k = blockIdx.x * blockDim.x + threadIdx.x;
  if (k >= N_BOX) return;
  float d  = dirs[k];
  float a1 = 0.f, a2 = 0.f;
  int idx = indices[k];
  if (idx > 0) {
    unsigned long long pk = refpack[idx - 1];
    unsigned jr = 0xFFFFFFFFu - (unsigned)(pk & 0xFFFFFFFFull);
    float ref = dirs[jr];
    float dd  = fabsf(d - ref);
    if (dd > PI_F) dd = 2.f * PI_F - dd;
    float sgn = (dd > 0.5f * PI_F) ? -1.f : 1.f;  // (1 - 2*gt)
    a1 = scores[k];
    a2 = a1 * sgn;
  }
  colsA[k] = make_float2(a1, a2);
  float* B = Bmat + (size_t)k * 16;
  B[0] = boxes[k * 7 + 0]; B[1] = boxes[k * 7 + 1]; B[2] = boxes[k * 7 + 2];
  B[3] = boxes[k * 7 + 3]; B[4] = boxes[k * 7 + 4]; B[5] = boxes[k * 7 + 5];
  B[6] = sinf(d); B[7] = cosf(d); B[8] = 1.f;
  B[9] = B[10] = B[11] = B[12] = B[13] = B[14] = B[15] = 0.f;
}

// ----------------- WMMA fusion: D = s_m @ B  and  D2 = (s_m*sign) @ B
// One wave per 16-cluster tile. B panel (4096x16 f32 = 256 KB) staged in
// LDS by the Tensor Data Mover once per workgroup.
__global__ void __launch_bounds__(128)
wmma_fuse_kernel(const int* __restrict__ indices,
                 const float2* __restrict__ colsA,
                 const float* __restrict__ Bmat,
                 float* __restrict__ acc1o, float* __restrict__ acc2o) {
  extern __shared__ float ldsB[];  // 4096 * 16 floats

  if (threadIdx.x < 32) {  // wave 0 issues the TDM descriptor
    unsigned long long ga = (unsigned long long)(uintptr_t)Bmat;
    u32x4 g0;
    g0[0] = 1u;                                   // count=1, user descriptor
    g0[1] = 0u;                                   // lds_addr = 0 (dyn-LDS base)
    g0[2] = (unsigned)(ga & 0xFFFFFFFFull);       // global_addr[31:0]
    g0[3] = (unsigned)((ga >> 32) & 0x1FFFFFFull) | (2u << 30);  // addr[56:32] | type=2
    i32x8 g1;
    g1[0] = (int)(2u << 16);      // data_size = 4B
    g1[1] = (int)(16u << 16);     // tensor_dim0 = 16
    g1[2] = (int)(4096u << 16);   // tensor_dim1 = 4096
    g1[3] = (int)(16u << 16);     // tile_dim0 = 16
    g1[4] = 4096;                 // tile_dim1 = 4096
    g1[5] = 16;                   // tensor_dim0_stride = 16
    g1[6] = 0; g1[7] = 0;
    i32x4 z4 = {0, 0, 0, 0};
#if __clang_major__ >= 23
    i32x8 z8 = {0, 0, 0, 0, 0, 0, 0, 0};
    __builtin_amdgcn_tensor_load_to_lds(g0, g1, z4, z4, z8, 0);
#else
    __builtin_amdgcn_tensor_load_to_lds(g0, g1, z4, z4, 0);
#endif
    __builtin_amdgcn_s_wait_tensorcnt(0);
  }
  __syncthreads();

  int wave = threadIdx.x >> 5, lane = threadIdx.x & 31;
  int rowbase = (blockIdx.x * 4 + wave) * 16;
  int half = lane >> 4;        // K-half select (A/B layout)
  int m    = lane & 15;        // A: row M; B/D: column N
  int want = rowbase + m + 1;  // cluster id this lane's A-row selects

  v8f acc1 = {};
  v8f acc2 = {};
  for (int k0 = 0; k0 < N_BOX; k0 += 4) {
    int ka = k0 + 2 * half;
    int i0 = indices[ka], i1 = indices[ka + 1];
    float2 c0 = colsA[ka], c1 = colsA[ka + 1];
    v2f A1, A2, B;
    A1[0] = (i0 == want) ? c0.x : 0.f;  A1[1] = (i1 == want) ? c1.x : 0.f;
    A2[0] = (i0 == want) ? c0.y : 0.f;  A2[1] = (i1 == want) ? c1.y : 0.f;
    B[0]  = ldsB[ka * 16 + m];
    B[1]  = ldsB[(ka + 1) * 16 + m];
    acc1 = __builtin_amdgcn_wmma_f32_16x16x4_f32(false, A1, false, B, (short)0,
                                                 acc1, false, false);
    acc2 = __builtin_amdgcn_wmma_f32_16x16x4_f32(false, A2, false, B, (short)0,
                                                 acc2, false, false);
  }
  // D layout: VGPR v -> row (rowbase + v + 8*half), col m
  for (int v = 0; v < 8; ++v) {
    int row = rowbase + v + 8 * half;
    acc1o[row * 16 + m] = acc1[v];
    acc2o[row * 16 + m] = acc2[v];
  }
}

// ---------------------------------------------- finalize rows -> outputs
__global__ void finalize_kernel(const float* __restrict__ acc1o,
                                const float* __restrict__ acc2o,
                                const float* __restrict__ s_sum,
                                const int* __restrict__ cnt,
                                const float* __restrict__ clscores,
                                int* __restrict__ validArr,
                                float* __restrict__ out) {
  int i = blockIdx.x * blockDim.x + threadIdx.x;
  if (i >= N_BOX) return;
  float ssum   = s_sum[i];
  bool  member = ssum > 0.f;
  float inv    = 1.f / fmaxf(ssum, 1e-12f);
  float b[7];
  for (int c = 0; c < 6; ++c) b[c] = acc1o[i * 16 + c] * inv;
  float S2 = acc2o[i * 16 + 6], C2 = acc2o[i * 16 + 7], G = acc2o[i * 16 + 8];
  // G = score_le - score_gt; if G >= 0 keep signed sums, else global flip
  float sgn = (G >= 0.f) ? 1.f : -1.f;
  b[6] = atan2f(sgn * S2, sgn * C2);

  float x = b[0], y = b[1], h = b[3], w = b[4], l = b[5], yaw = b[6];
  float cy_ = cosf(yaw), sy_ = sinf(yaw);
  const float xs[8] = {1, 1, -1, -1, 1, 1, -1, -1};
  const float ys[8] = {-1, 1, 1, -1, -1, 1, 1, -1};
  bool in_range = true;
  for (int c = 0; c < 8; ++c) {
    float cxl = 0.5f * l * xs[c], cyw = 0.5f * w * ys[c];
    float px = cxl * cy_ - cyw * sy_ + x;
    float py = cxl * sy_ + cyw * cy_ + y;
    in_range = in_range && (px > -140.8f) && (px < 140.8f) &&
               (py > -40.f) && (py < 40.f);
  }
  (void)h;
  bool valid  = member && in_range;
  validArr[i] = valid ? 1 : 0;

  // s_fused = min( sum_k sorted_desc(s)^k , 1 )
  float ls[64];
  int mct = cnt[i]; if (mct > 64) mct = 64;
  for (int s = 0; s < mct; ++s) ls[s] = clscores[i * 64 + s];
  for (int a = 1; a < mct; ++a) {
    float v = ls[a]; int p = a - 1;
    while (p >= 0 && ls[p] < v) { ls[p + 1] = ls[p]; --p; }
    ls[p + 1] = v;
  }
  float sf = 0.f;
  for (int s = 0; s < mct; ++s) sf += powf(ls[s], (float)(s + 1));
  sf = fminf(sf, 1.f);

  float* boxes_out  = out;
  float* scores_out = out + N_BOX * 7;
  float* valid_out  = out + N_BOX * 8;
  for (int c = 0; c < 7; ++c) boxes_out[i * 7 + c] = valid ? b[c] : 0.f;
  scores_out[i] = valid ? sf : 0.f;
  valid_out[i]  = valid ? 1.f : 0.f;
}

// ---------------------------------------- cumsum(valid) -> new_indices
__global__ void scan_kernel(const int* __restrict__ validArr,
                            const int* __restrict__ indices,
                            float* __restrict__ out) {
  __shared__ int part[1024];
  __shared__ int snew[N_BOX];
  int t = threadIdx.x;
  int v0 = validArr[4 * t], v1 = validArr[4 * t + 1];
  int v2 = validArr[4 * t + 2], v3 = validArr[4 * t + 3];
  int p0 = v0, p1 = p0 + v1, p2 = p1 + v2, p3 = p2 + v3;
  part[t] = p3;
  __syncthreads();
  for (int off = 1; off < 1024; off <<= 1) {
    int add = (t >= off) ? part[t - off] : 0;
    __syncthreads();
    part[t] += add;
    __syncthreads();
  }
  int base = part[t] - p3;
  snew[4 * t + 0] = base + p0; snew[4 * t + 1] = base + p1;
  snew[4 * t + 2] = base + p2; snew[4 * t + 3] = base + p3;
  __syncthreads();
  float* nidx_out = out + N_BOX * 9;
  for (int b = 0; b < 4; ++b) {
    int j = 4 * t + b;
    int idx = indices[j];
    int val = 0;
    if (idx > 0) { int r = idx - 1; if (validArr[r]) val = snew[r]; }
    nidx_out[j] = (float)val;
  }
}

// ------------------------------------------------------------ launcher
extern "C" void kernel_launch(void* const* d_in, const int* in_sizes, int n_in,
                              void* d_out, int out_size, void* d_ws, size_t ws_size,
                              hipStream_t stream) {
  (void)in_sizes; (void)n_in; (void)out_size; (void)ws_size;
  const float* boxes  = (const float*)d_in[0];
  const float* scores = (const float*)d_in[1];
  float* out = (float*)d_out;

  char* ws = (char*)d_ws;
  size_t off = 0;
  auto alloc = [&](size_t bytes) -> void* {
    void* p = ws + off;
    off += (bytes + 255) & ~(size_t)255;
    return p;
  };
  unsigned long long* refpack = (unsigned long long*)alloc(N_BOX * 8);
  uint32_t* mask  = (uint32_t*)alloc((size_t)N_BOX * 128 * 4);  // 2 MB bitmask
  float*  dirs    = (float*)alloc(N_BOX * 4);
  int*    indices = (int*)alloc(N_BOX * 4);
  float*  s_sum   = (float*)alloc(N_BOX * 4);
  int*    cnt     = (int*)alloc(N_BOX * 4);
  float*  clsc    = (float*)alloc((size_t)N_BOX * 64 * 4);      // 1 MB
  float2* colsA   = (float2*)alloc(N_BOX * 8);
  float*  Bmat    = (float*)alloc((size_t)N_BOX * 16 * 4);      // 256 KB panel
  float*  acc1o   = (float*)alloc((size_t)N_BOX * 16 * 4);
  float*  acc2o   = (float*)alloc((size_t)N_BOX * 16 * 4);
  int*    validA  = (int*)alloc(N_BOX * 4);

  prep0_kernel<<<N_BOX / 256, 256, 0, stream>>>(boxes, dirs, s_sum, cnt, refpack);
  iou_mask_kernel<<<N_BOX, 128, 0, stream>>>(boxes, mask);
  assign_kernel<<<1, 1024, 0, stream>>>(mask, indices);
  stats_kernel<<<N_BOX / 256, 256, 0, stream>>>(scores, indices, s_sum, cnt, clsc, refpack);
  prep2_kernel<<<N_BOX / 256, 256, 0, stream>>>(boxes, scores, dirs, indices, refpack, colsA, Bmat);
  wmma_fuse_kernel<<<64, 128, (size_t)N_BOX * 16 * 4, stream>>>(indices, colsA, Bmat, acc1o, acc2o);
  finalize_kernel<<<N_BOX / 256, 256, 0, stream>>>(acc1o, acc2o, s_sum, cnt, clsc, validA, out);
  scan_kernel<<<1, 1024, 0, stream>>>(validA, indices, out);
}